// ExpanderGIN_77094662963228
// MI455X (gfx1250) — compile-verified
//
#include <hip/hip_runtime.h>

typedef __attribute__((ext_vector_type(2))) float v2f;
typedef __attribute__((ext_vector_type(8))) float v8f;

#define N_NODES 100000
#define N_EDGES 625000
#define DIM     128

// ---------------- zero the aggregation scratch ----------------
__global__ void gin_zero(float* __restrict__ agg, int n) {
  int i = blockIdx.x * blockDim.x + threadIdx.x;
  int stride = gridDim.x * blockDim.x;
  for (; i < n; i += stride) agg[i] = 0.0f;
}

// ---------------- edge scatter-add: agg[dst] += x[src] ----------------
// 32 threads per edge; each thread handles 4 consecutive features (float4).
// Edge index is wave-uniform; per-edge traffic is one coalesced 512B gather
// + 128 float atomics (global_atomic_add_f32), all L2-resident (x+agg = 102MB < 192MB L2).
__global__ void __launch_bounds__(256) gin_scatter(const float* __restrict__ x,
                                                   const int* __restrict__ src,
                                                   const int* __restrict__ dst,
                                                   float* __restrict__ agg) {
  long long i = (long long)blockIdx.x * blockDim.x + threadIdx.x;
  int e = (int)(i >> 5);           // edge id (uniform across a wave)
  int c = ((int)i & 31) << 2;      // feature base: 0,4,...,124
  int s = src[e];
  int d = dst[e];
  const float4 v = *(const float4*)(x + (long long)s * DIM + c);
  float* a = agg + (long long)d * DIM + c;
  unsafeAtomicAdd(a + 0, v.x);
  unsafeAtomicAdd(a + 1, v.y);
  unsafeAtomicAdd(a + 2, v.z);
  unsafeAtomicAdd(a + 3, v.w);
}

// ---------------- fused (x + agg) @ W^T + b, ReLU, via WMMA f32 ----------------
// One wave computes one 16x16 output tile. K=128 in 32 steps of V_WMMA_F32_16X16X4_F32.
__global__ void __launch_bounds__(256) gin_gemm(const float* __restrict__ x,
                                                const float* __restrict__ agg,
                                                const float* __restrict__ W,
                                                const float* __restrict__ bias,
                                                float* __restrict__ out) {
  const int lane   = threadIdx.x & 31;
  const int wave   = threadIdx.x >> 5;
  const int waveId = blockIdx.x * 8 + wave;
  const int mtile  = waveId >> 3;   // 0..6249  (node tile)
  const int ntile  = waveId & 7;    // 0..7     (output-feature tile)
  const int laneLo = lane & 15;
  const int laneHi = lane >> 4;     // 0 or 1

  // A: rows of h = x + agg ; B: rows of W (== columns of W^T)
  const float* xr = x   + (long long)(mtile * 16 + laneLo) * DIM;
  const float* ar = agg + (long long)(mtile * 16 + laneLo) * DIM;
  const float* wr = W   + (long long)(ntile * 16 + laneLo) * DIM;

  v8f c = {};
#pragma unroll
  for (int k0 = 0; k0 < DIM; k0 += 4) {
    const int k = k0 + 2 * laneHi;            // ISA A/B f32 layout: K split by lane half
    v2f xv = *(const v2f*)(xr + k);
    v2f av = *(const v2f*)(ar + k);
    v2f a  = xv + av;                         // fused h = x + agg
    v2f b  = *(const v2f*)(wr + k);
    // 8 args: (neg_a, A, neg_b, B, c_mod, C, reuse_a, reuse_b)
    c = __builtin_amdgcn_wmma_f32_16x16x4_f32(false, a, false, b,
                                              (short)0, c, false, false);
  }

  // C/D layout: VGPR r, lanes 0-15 -> M=r, N=lane ; lanes 16-31 -> M=r+8, N=lane-16
  const int n = ntile * 16 + laneLo;
  const float bv = bias[n];
#pragma unroll
  for (int r = 0; r < 8; ++r) {
    const int m = mtile * 16 + r + 8 * laneHi;
    float v = c[r] + bv;
    out[(long long)m * DIM + n] = v > 0.0f ? v : 0.0f;
  }
}

extern "C" void kernel_launch(void* const* d_in, const int* in_sizes, int n_in,
                              void* d_out, int out_size, void* d_ws, size_t ws_size,
                              hipStream_t stream) {
  const float* x  = (const float*)d_in[0];          // [N, 128]
  const int*   ei = (const int*)d_in[1];            // [2, E] row-major: src then dst
  const float* W  = (const float*)d_in[2];          // [128, 128] (out, in)
  const float* b  = (const float*)d_in[3];          // [128]
  float* out = (float*)d_out;                       // [N, 128]
  float* agg = (float*)d_ws;                        // [N, 128] scratch

  const int total = N_NODES * DIM;                  // 12.8M floats
  gin_zero<<<(total + 1023) / 1024, 1024, 0, stream>>>(agg, total);

  const long long sthreads = (long long)N_EDGES * 32;   // 20,000,000
  gin_scatter<<<(int)(sthreads / 256), 256, 0, stream>>>(x, ei, ei + N_EDGES, agg);

  const int waves = (N_NODES / 16) * (DIM / 16);    // 50,000 tiles, 8 waves/block
  gin_gemm<<<waves / 8, 256, 0, stream>>>(x, agg, W, b, out);
}